// HigherOrderMixHop_36369783062757
// MI455X (gfx1250) — compile-verified
//
#include <hip/hip_runtime.h>

typedef float v2f __attribute__((ext_vector_type(2)));
typedef float v8f __attribute__((ext_vector_type(8)));

// ---------------------------------------------------------------------------
// Dense GEMM via V_WMMA_F32_16X16X4_F32 (full fp32, matches reference math).
// out[M x NC] = act(A[M x 128] @ W[128 x NC] + bias[NC]) (+ res[M x NC])
//
// Block: 256 threads = 8 waves. Each wave computes a 32-row strip
// (2 M-tiles x NT N-tiles), so every B fragment read from LDS feeds 2 WMMAs.
//
// W is staged in LDS pre-paired: sW2[k2*NC + c] = {W[2*k2][c], W[2*k2+1][c]}
// so a lane's B fragment (rows ka, ka+1 at column c) is ONE aligned
// ds_load_b64 straight into an even VGPR pair (no repack movs).
//
// Fragment layouts per CDNA5 ISA 7.12.2:
//   A (16x4 MxK):  lanes 0-15 -> K=0,1 ; lanes 16-31 -> K=2,3
//   B (4x16 KxN):  lane -> N col ; VGPR v: lanes 0-15 K=v, lanes 16-31 K=v+2
//   C/D (16x16):   lane -> N col ; VGPR v: lanes 0-15 M=v, lanes 16-31 M=v+8
// ---------------------------------------------------------------------------
template <int NC, bool RELU, bool RES>
__global__ __launch_bounds__(256) void gemm_wmma(
    const float* __restrict__ A, const float* __restrict__ W,
    const float* __restrict__ bias, const float* __restrict__ res,
    float* __restrict__ out, int nrows) {
  __shared__ v2f sW2[64 * NC];  // NC=128 -> 64 KB
  for (int i = threadIdx.x; i < 64 * NC; i += 256) {
    const int k2 = i / NC;
    const int col = i - k2 * NC;
    v2f w;
    w.x = W[(2 * k2) * NC + col];
    w.y = W[(2 * k2 + 1) * NC + col];
    sW2[i] = w;
  }
  __syncthreads();

  const int wave = threadIdx.x >> 5;
  const int lane = threadIdx.x & 31;
  const int half = lane >> 4;  // 0: lanes 0-15, 1: lanes 16-31
  const int l16 = lane & 15;
  const int m0 = (blockIdx.x * 8 + wave) * 32;
  if (m0 >= nrows) return;

  constexpr int NT = NC / 16;
  v8f acc[2][NT];
  const v8f vzero = {};
#pragma unroll
  for (int mt = 0; mt < 2; ++mt)
#pragma unroll
    for (int t = 0; t < NT; ++t) acc[mt][t] = vzero;

  // Per-lane A row pointers (clamped: tail tile reads stay in-bounds; the
  // corresponding stores are guarded below). Lane's pair for K-step s lives
  // at float2 index (2*s + half) of its row.
  const int r0 = min(m0 + l16, nrows - 1);
  const int r1 = min(m0 + 16 + l16, nrows - 1);
  const v2f* __restrict__ a0 = (const v2f*)(A + (size_t)r0 * 128);
  const v2f* __restrict__ a1 = (const v2f*)(A + (size_t)r1 * 128);

  v2f af0 = a0[half];
  v2f af1 = a1[half];

#pragma unroll 4
  for (int s = 0; s < 32; ++s) {
    const v2f ac0 = af0;
    const v2f ac1 = af1;
    if (s < 31) {  // software prefetch next A fragments
      af0 = a0[2 * (s + 1) + half];
      af1 = a1[2 * (s + 1) + half];
    }
    const v2f* __restrict__ brow = sW2 + (size_t)(2 * s + half) * NC + l16;
#pragma unroll
    for (int t = 0; t < NT; ++t) {
      const v2f bf = brow[t * 16];
      acc[0][t] = __builtin_amdgcn_wmma_f32_16x16x4_f32(
          false, ac0, false, bf, (short)0, acc[0][t], false, false);
      acc[1][t] = __builtin_amdgcn_wmma_f32_16x16x4_f32(
          false, ac1, false, bf, (short)0, acc[1][t], false, false);
    }
  }

#pragma unroll
  for (int mt = 0; mt < 2; ++mt) {
    const int mbase = m0 + mt * 16;
    if (mbase >= nrows) break;  // nrows is a multiple of 16
#pragma unroll
    for (int t = 0; t < NT; ++t) {
      const int col = t * 16 + l16;
      const float bv = bias[col];
#pragma unroll
      for (int v = 0; v < 8; ++v) {
        const int row = mbase + v + half * 8;
        float x = acc[mt][t][v] + bv;
        if (RELU) x = fmaxf(x, 0.f);
        if (RES) x += res[(size_t)row * NC + col];
        out[(size_t)row * NC + col] = x;
      }
    }
  }
}

// ---------------------------------------------------------------------------
// SpMM (COO): out[dst,:] += val * x[src,:], D = 128. One wave per edge
// (index forced wave-uniform -> scalar idx/val loads), lane handles a float4
// -> global_load_b128 gather (L2-resident) + 4 global_atomic_add_f32 scatter.
// ---------------------------------------------------------------------------
__global__ __launch_bounds__(256) void spmm_atomic(
    const int* __restrict__ idx, const float* __restrict__ val,
    const float* __restrict__ x, float* __restrict__ out, int nedges) {
  const int e =
      __builtin_amdgcn_readfirstlane(blockIdx.x * 8 + (threadIdx.x >> 5));
  const int lane = threadIdx.x & 31;
  if (e >= nedges) return;
  const int dst = idx[e];
  const int src = idx[nedges + e];
  const float v = val[e];
  const float4 d = ((const float4*)(x + (size_t)src * 128))[lane];
  float* o = out + (size_t)dst * 128 + lane * 4;
  unsafeAtomicAdd(o + 0, v * d.x);
  unsafeAtomicAdd(o + 1, v * d.y);
  unsafeAtomicAdd(o + 2, v * d.z);
  unsafeAtomicAdd(o + 3, v * d.w);
}

// ---------------------------------------------------------------------------
// Attention: scores_k = q . key_k (DP=64), softmax over 5, comb = sum w_k f_k.
// One wave per node; cross-lane reduce via shfl_xor (wave32).
// ---------------------------------------------------------------------------
__global__ __launch_bounds__(256) void attn_combine(
    const float* __restrict__ q, const float* __restrict__ k0,
    const float* __restrict__ k1, const float* __restrict__ k2,
    const float* __restrict__ k3, const float* __restrict__ k4,
    const float* __restrict__ f0, const float* __restrict__ f1,
    const float* __restrict__ f2, const float* __restrict__ f3,
    const float* __restrict__ f4, float* __restrict__ comb, int n) {
  const int node = blockIdx.x * 8 + (threadIdx.x >> 5);
  const int lane = threadIdx.x & 31;
  if (node >= n) return;

  const float2 qv = ((const float2*)(q + (size_t)node * 64))[lane];
  const float* ks[5] = {k0, k1, k2, k3, k4};
  float sc[5];
#pragma unroll
  for (int k = 0; k < 5; ++k) {
    const float2 kv = ((const float2*)(ks[k] + (size_t)node * 64))[lane];
    float s = qv.x * kv.x + qv.y * kv.y;
#pragma unroll
    for (int m = 16; m >= 1; m >>= 1) s += __shfl_xor(s, m, 32);
    sc[k] = s;
  }
  float mx = sc[0];
#pragma unroll
  for (int k = 1; k < 5; ++k) mx = fmaxf(mx, sc[k]);
  float sum = 0.f;
#pragma unroll
  for (int k = 0; k < 5; ++k) {
    sc[k] = __expf(sc[k] - mx);
    sum += sc[k];
  }
  const float inv = 1.f / sum;

  const float* fs[5] = {f0, f1, f2, f3, f4};
  float cx = 0.f, cy = 0.f, cz = 0.f, cw = 0.f;
#pragma unroll
  for (int k = 0; k < 5; ++k) {
    const float w = sc[k] * inv;
    const float4 fv = ((const float4*)(fs[k] + (size_t)node * 128))[lane];
    cx += w * fv.x;
    cy += w * fv.y;
    cz += w * fv.z;
    cw += w * fv.w;
  }
  float4 c;
  c.x = cx; c.y = cy; c.z = cz; c.w = cw;
  ((float4*)(comb + (size_t)node * 128))[lane] = c;
}

// ---------------------------------------------------------------------------
// Host orchestration (all async, graph-capture safe).
// Input order = setup_inputs() dict order.
// ---------------------------------------------------------------------------
extern "C" void kernel_launch(void* const* d_in, const int* in_sizes, int n_in,
                              void* d_out, int out_size, void* d_ws,
                              size_t ws_size, hipStream_t stream) {
  (void)n_in; (void)out_size; (void)ws_size;
  const float* features = (const float*)d_in[0];
  const float* edge_features = (const float*)d_in[1];
  const int* adj_idx[4] = {(const int*)d_in[2], (const int*)d_in[4],
                           (const int*)d_in[6], (const int*)d_in[8]};
  const float* adj_val[4] = {(const float*)d_in[3], (const float*)d_in[5],
                             (const float*)d_in[7], (const float*)d_in[9]};
  const float* Wt = (const float*)d_in[10];
  const float* fo_W1 = (const float*)d_in[11];
  const float* fo_W2 = (const float*)d_in[12];
  const float* so_W1 = (const float*)d_in[13];
  const float* so_W2 = (const float*)d_in[14];
  const float* to_W1 = (const float*)d_in[15];
  const float* to_W2 = (const float*)d_in[16];
  const float* q4_W1 = (const float*)d_in[17];
  const float* q4_W2 = (const float*)d_in[18];
  const float* edge_W1 = (const float*)d_in[19];
  const float* edge_W2 = (const float*)d_in[20];
  const float* out_W1 = (const float*)d_in[21];
  const float* out_W2 = (const float*)d_in[22];
  const float* bt = (const float*)d_in[23];
  const float* fo_b1 = (const float*)d_in[24];
  const float* fo_b2 = (const float*)d_in[25];
  const float* so_b1 = (const float*)d_in[26];
  const float* so_b2 = (const float*)d_in[27];
  const float* to_b1 = (const float*)d_in[28];
  const float* to_b2 = (const float*)d_in[29];
  const float* q4_b1 = (const float*)d_in[30];
  const float* q4_b2 = (const float*)d_in[31];
  const float* edge_b1 = (const float*)d_in[32];
  const float* edge_b2 = (const float*)d_in[33];
  const float* out_b1 = (const float*)d_in[34];
  const float* out_b2 = (const float*)d_in[35];
  const float* Wq = (const float*)d_in[36];
  const float* Wk = (const float*)d_in[37];
  const float* bq = (const float*)d_in[38];
  const float* bk = (const float*)d_in[39];

  const int n = in_sizes[0] / 128;
  const size_t ND = (size_t)n * 128;
  const size_t NH = (size_t)n * 64;

  float* ws = (float*)d_ws;
  float* h = ws;                 // feature transform
  float* ef = ws + ND;           // edge-MLP output; reused later as comb
  float* tmp = ws + 2 * ND;      // MLP hidden
  float* s[4] = {ws + 3 * ND, ws + 4 * ND, ws + 5 * ND, ws + 6 * ND};
  float* qb = ws + 7 * ND;
  float* keys[5] = {qb + NH, qb + 2 * NH, qb + 3 * NH, qb + 4 * NH,
                    qb + 5 * NH};
  float* comb = ef;

  const dim3 blk(256);
  const int gemmGrid = (n + 255) / 256;  // 8 waves x 32 rows per block
  const int attnGrid = (n + 7) / 8;

  // Zero the 4 SpMM accumulators (only buffers that need init).
  hipMemsetAsync(s[0], 0, 4 * ND * sizeof(float), stream);

  // h = features @ Wt + bt
  gemm_wmma<128, false, false><<<gemmGrid, blk, 0, stream>>>(
      features, Wt, bt, nullptr, h, n);

  // ef = MLP2(edge_features)
  gemm_wmma<128, true, false><<<gemmGrid, blk, 0, stream>>>(
      edge_features, edge_W1, edge_b1, nullptr, tmp, n);
  gemm_wmma<128, false, false><<<gemmGrid, blk, 0, stream>>>(
      tmp, edge_W2, edge_b2, nullptr, ef, n);

  // sK = spmm(adjK, h)
  for (int a = 0; a < 4; ++a) {
    const int E = in_sizes[3 + 2 * a];
    spmm_atomic<<<(E + 7) / 8, blk, 0, stream>>>(adj_idx[a], adj_val[a], h,
                                                 s[a], E);
  }

  // a1 = MLP2(s1) + spmm(adj1, ef)   (a1 aliases s1)
  gemm_wmma<128, true, false><<<gemmGrid, blk, 0, stream>>>(
      s[0], fo_W1, fo_b1, nullptr, tmp, n);
  gemm_wmma<128, false, false><<<gemmGrid, blk, 0, stream>>>(
      tmp, fo_W2, fo_b2, nullptr, s[0], n);
  {
    const int E = in_sizes[3];
    spmm_atomic<<<(E + 7) / 8, blk, 0, stream>>>(adj_idx[0], adj_val[0], ef,
                                                 s[0], E);
  }

  // aK = MLP2(sK) + sK for K = 2..4 (in-place, residual = old sK)
  const float* W1s[3] = {so_W1, to_W1, q4_W1};
  const float* b1s[3] = {so_b1, to_b1, q4_b1};
  const float* W2s[3] = {so_W2, to_W2, q4_W2};
  const float* b2s[3] = {so_b2, to_b2, q4_b2};
  for (int o = 0; o < 3; ++o) {
    gemm_wmma<128, true, false><<<gemmGrid, blk, 0, stream>>>(
        s[o + 1], W1s[o], b1s[o], nullptr, tmp, n);
    gemm_wmma<128, false, true><<<gemmGrid, blk, 0, stream>>>(
        tmp, W2s[o], b2s[o], s[o + 1], s[o + 1], n);
  }

  // q = h @ Wq + bq ; key_k = f_k @ Wk + bk
  gemm_wmma<64, false, false><<<gemmGrid, blk, 0, stream>>>(h, Wq, bq, nullptr,
                                                            qb, n);
  const float* fsets[5] = {h, s[0], s[1], s[2], s[3]};
  for (int k = 0; k < 5; ++k)
    gemm_wmma<64, false, false><<<gemmGrid, blk, 0, stream>>>(
        fsets[k], Wk, bk, nullptr, keys[k], n);

  // softmax-attention combine
  attn_combine<<<attnGrid, blk, 0, stream>>>(
      qb, keys[0], keys[1], keys[2], keys[3], keys[4], h, s[0], s[1], s[2],
      s[3], comb, n);

  // out = MLP2(comb)
  gemm_wmma<128, true, false><<<gemmGrid, blk, 0, stream>>>(
      comb, out_W1, out_b1, nullptr, tmp, n);
  gemm_wmma<128, false, false><<<gemmGrid, blk, 0, stream>>>(
      tmp, out_W2, out_b2, nullptr, (float*)d_out, n);
}